// GIN_64544768525161
// MI455X (gfx1250) — compile-verified
//
#include <hip/hip_runtime.h>
#include <hip/hip_bf16.h>

typedef __attribute__((ext_vector_type(2))) float v2f;
typedef __attribute__((ext_vector_type(4))) float v4f;
typedef __attribute__((ext_vector_type(8))) float v8f;

#define N_NODES 100000
#define N_EDGES 1600000
#define NGRAPH  1024
#define IN_F    32
#define H_F     64
#define EV_F    16

// ---------------------------------------------------------------------------
// Edge aggregation: AGG[dst[e]][f..f+3] += H[src[e]][f..f+3].
// One thread per (edge, 4-feature chunk): b128 coalesced loads, 4 f32 atomics
// resolved in L2. This phase is the HBM/atomic roofline of the whole net.
// ---------------------------------------------------------------------------
__global__ void gin_edge_agg(const float* __restrict__ H,
                             const int* __restrict__ src,
                             const int* __restrict__ dst,
                             float* __restrict__ AGG,
                             int nEdges, int logC) {  // logC = log2(F/4)
  const int C = 1 << logC;
  int idx = blockIdx.x * blockDim.x + threadIdx.x;
  if (idx >= (nEdges << logC)) return;
  const int e = idx >> logC;
  const int c = idx & (C - 1);
  const int s = src[e];
  const int d = dst[e];
  const v4f h = *(const v4f*)(H + (((size_t)s << logC) + c) * 4);
  float* p = AGG + (((size_t)d << logC) + c) * 4;
  atomicAdd(p + 0, h.x);
  atomicAdd(p + 1, h.y);
  atomicAdd(p + 2, h.z);
  atomicAdd(p + 3, h.w);
}

// ---------------------------------------------------------------------------
// Generic fp32 WMMA linear layer:  Y[R,N] = act(X[R,K] @ W[K,N] + b[N])
// One wave computes a 16 x (16*NT) output tile with V_WMMA_F32_16X16X4_F32.
// Per k-step: ONE load clause fills the A frag + all NT B frags, then NT
// WMMAs issue back-to-back (single loadcnt wait per burst; the scheduler can
// hoist the next iteration's clause above the current WMMA burst).
//   A frag (16x4 f32): lanes 0-15 -> M=lane, K=k,k+1 ; lanes 16-31 -> K=k+2,k+3
//   B frag (4x16 f32): lanes 0-15 -> N=lane, K=k,k+1 ; lanes 16-31 -> K=k+2,k+3
//   C/D: VGPR j -> row (j + 8*(lane>=16)), col = lane&15
// OOB rows are CLAMPED (not zeroed): output row m depends only on A row m and
// OOB rows are masked at the store, so the K loop has no guarded loads.
// ---------------------------------------------------------------------------
template <int NT>
__global__ void linear_wmma(const float* __restrict__ X,
                            const float* __restrict__ W,
                            const float* __restrict__ Bv,
                            float* __restrict__ Y,
                            int R, int K, int N, int doRelu) {
  const int lane = threadIdx.x & 31;
  const int wave = threadIdx.x >> 5;
  const int wavesPerBlock = blockDim.x >> 5;
  const int rowTiles = (R + 15) >> 4;
  const int colGroups = N / (16 * NT);
  const int tile = blockIdx.x * wavesPerBlock + wave;
  if (tile >= rowTiles * colGroups) return;
  const int rt = tile % rowTiles;
  const int ct = tile / rowTiles;
  const int row0 = rt << 4;
  const int n0 = ct * (16 * NT);
  const int m = lane & 15;          // row (A) / col (B,C,D) within a 16-tile
  const int kh = (lane >> 4) << 1;  // K sub-offset: 0 (lanes 0-15) or 2

  const int r = min(row0 + m, R - 1);                       // row clamp
  const v2f* __restrict__ pA = (const v2f*)(X + (size_t)r * K + kh);
  const float* __restrict__ pB0 = W + (size_t)kh * N + n0 + m;
  const float* __restrict__ pB1 = pB0 + N;
  const size_t strideB = (size_t)4 * N;

  v8f acc[NT];
#pragma unroll
  for (int t = 0; t < NT; ++t) acc[t] = (v8f){};

#pragma unroll 2
  for (int k = 0; k < K; k += 4) {
    // ---- load phase: one clause, no WMMA interleaved ----
    const v2f a = *pA;                 // global_load_b64, 8B aligned
    v2f b[NT];
#pragma unroll
    for (int t = 0; t < NT; ++t) {
      b[t].x = pB0[t * 16];
      b[t].y = pB1[t * 16];
    }
    // ---- MMA burst: NT back-to-back WMMAs, all operands resident ----
#pragma unroll
    for (int t = 0; t < NT; ++t) {
      acc[t] = __builtin_amdgcn_wmma_f32_16x16x4_f32(
          /*neg_a=*/false, a, /*neg_b=*/false, b[t],
          /*c_mod=*/(short)0, acc[t], /*reuse_a=*/false, /*reuse_b=*/false);
    }
    pA += 2;            // +4 floats
    pB0 += strideB;     // +4 rows of W
    pB1 += strideB;
  }

  float bias[NT];
#pragma unroll
  for (int t = 0; t < NT; ++t) bias[t] = Bv[n0 + t * 16 + m];

  const int rbase = row0 + ((lane >> 4) << 3);
  float* __restrict__ py = Y + (size_t)rbase * N + n0 + m;
  if (row0 + 16 <= R) {               // wave-uniform fast path: full tile
#pragma unroll
    for (int j = 0; j < 8; ++j) {
#pragma unroll
      for (int t = 0; t < NT; ++t) {
        float v = acc[t][j] + bias[t];
        if (doRelu) v = fmaxf(v, 0.0f);
        py[t * 16] = v;
      }
      py += N;
    }
  } else {                            // ragged last row-tile
#pragma unroll
    for (int j = 0; j < 8; ++j) {
      if (rbase + j < R) {
#pragma unroll
        for (int t = 0; t < NT; ++t) {
          float v = acc[t][j] + bias[t];
          if (doRelu) v = fmaxf(v, 0.0f);
          py[t * 16] = v;
        }
      }
      py += N;
    }
  }
}

static inline int cdiv(long long a, long long b) { return (int)((a + b - 1) / b); }

static void run_linear(const float* X, const float* W, const float* b, float* Y,
                       int R, int K, int N, int relu, hipStream_t stream) {
  const int BT = 256;  // 8 waves
  if (N % 64 == 0) {
    int tiles = cdiv(R, 16) * (N / 64);
    linear_wmma<4><<<cdiv(tiles, 8), BT, 0, stream>>>(X, W, b, Y, R, K, N, relu);
  } else {  // N == 32
    int tiles = cdiv(R, 16) * (N / 32);
    linear_wmma<2><<<cdiv(tiles, 8), BT, 0, stream>>>(X, W, b, Y, R, K, N, relu);
  }
}

// ---------------------------------------------------------------------------
// Per-graph pooling accumulation: sums[gid[n]][f] += H[n][f]; counts once.
// ---------------------------------------------------------------------------
__global__ void pool_accum(const float* __restrict__ H,
                           const int* __restrict__ gid,
                           float* __restrict__ sums,
                           float* __restrict__ cnt,
                           int nNodes, int logF, int doCnt) {
  const int F = 1 << logF;
  int idx = blockIdx.x * blockDim.x + threadIdx.x;
  if (idx >= (nNodes << logF)) return;
  const int node = idx >> logF;
  const int f = idx & (F - 1);
  const int g = gid[node];
  atomicAdd(&sums[((size_t)g << logF) + f], H[idx]);
  if (doCnt && f == 0) atomicAdd(&cnt[g], 1.0f);
}

// ---------------------------------------------------------------------------
// hg = (pool(h1) + 2*pool(h2)) / max(cnt,1), concat event feats -> [B, 80]
// ---------------------------------------------------------------------------
__global__ void finalize_hg(const float* __restrict__ sum1,
                            const float* __restrict__ sum2,
                            const float* __restrict__ cnt,
                            const float* __restrict__ ev,
                            float* __restrict__ tot) {
  int idx = blockIdx.x * blockDim.x + threadIdx.x;
  if (idx >= NGRAPH * (H_F + EV_F)) return;
  const int g = idx / (H_F + EV_F);
  const int f = idx % (H_F + EV_F);
  float v;
  if (f < H_F) {
    const float c = fmaxf(cnt[g], 1.0f);
    v = (sum1[g * H_F + f] + 2.0f * sum2[g * H_F + f]) / c;
  } else {
    v = ev[g * EV_F + (f - H_F)];
  }
  tot[idx] = v;
}

// ---------------------------------------------------------------------------
// Final layer: [1024,32] @ [32,1] + b  (trivially small)
// ---------------------------------------------------------------------------
__global__ void final_dot(const float* __restrict__ X,
                          const float* __restrict__ W,
                          const float* __restrict__ Bv,
                          float* __restrict__ Y, int R, int K) {
  int r = blockIdx.x * blockDim.x + threadIdx.x;
  if (r >= R) return;
  float s = Bv[0];
#pragma unroll 8
  for (int k = 0; k < K; ++k) s += X[(size_t)r * K + k] * W[k];
  Y[r] = s;
}

extern "C" void kernel_launch(void* const* d_in, const int* in_sizes, int n_in,
                              void* d_out, int out_size, void* d_ws, size_t ws_size,
                              hipStream_t stream) {
  const float* in_feat = (const float*)d_in[0];
  const float* ev      = (const float*)d_in[1];
  const int*   src     = (const int*)d_in[2];
  const int*   dst     = (const int*)d_in[3];
  const int*   gid     = (const int*)d_in[4];
  const float* c0_w1 = (const float*)d_in[5];
  const float* c0_b1 = (const float*)d_in[6];
  const float* c0_w2 = (const float*)d_in[7];
  const float* c0_b2 = (const float*)d_in[8];
  const float* c1_w1 = (const float*)d_in[9];
  const float* c1_b1 = (const float*)d_in[10];
  const float* c1_w2 = (const float*)d_in[11];
  const float* c1_b2 = (const float*)d_in[12];
  const float* lw[7], *lb[7];
  for (int i = 0; i < 7; ++i) { lw[i] = (const float*)d_in[13 + 2 * i]; lb[i] = (const float*)d_in[14 + 2 * i]; }
  float* out = (float*)d_out;

  // Workspace layout (floats)
  float* w = (float*)d_ws;
  float* AGG0 = w;                 w += (size_t)N_NODES * IN_F;
  float* T1   = w;                 w += (size_t)N_NODES * H_F;
  float* H1   = w;                 w += (size_t)N_NODES * H_F;
  float* AGG1 = w;                 w += (size_t)N_NODES * H_F;
  float* T2   = w;                 w += (size_t)N_NODES * H_F;
  float* H2   = w;                 w += (size_t)N_NODES * H_F;
  float* SUM1 = w;                 w += (size_t)NGRAPH * H_F;
  float* SUM2 = w;                 w += (size_t)NGRAPH * H_F;
  float* CNT  = w;                 w += NGRAPH;
  float* TOT0 = w;                 w += (size_t)NGRAPH * (H_F + EV_F);
  float* TA   = w;                 w += (size_t)NGRAPH * 256;
  float* TB   = w;                 w += (size_t)NGRAPH * 256;

  const int BT = 256;  // 8 waves per block

  // ---- GIN layer 0 ----
  // x = in_feat + segment_sum(in_feat[src], dst): init AGG0 with self, add edges
  hipMemcpyAsync(AGG0, in_feat, (size_t)N_NODES * IN_F * sizeof(float),
                 hipMemcpyDeviceToDevice, stream);
  {
    long long n = (long long)N_EDGES * (IN_F / 4);
    gin_edge_agg<<<cdiv(n, BT), BT, 0, stream>>>(in_feat, src, dst, AGG0, N_EDGES, 3);
  }
  // T1 = relu(AGG0 @ c0_w1 + c0_b1); H1 = relu(T1 @ c0_w2 + c0_b2)
  run_linear(AGG0, c0_w1, c0_b1, T1, N_NODES, IN_F, H_F, 1, stream);
  run_linear(T1, c0_w2, c0_b2, H1, N_NODES, H_F, H_F, 1, stream);

  // ---- GIN layer 1 ----
  hipMemcpyAsync(AGG1, H1, (size_t)N_NODES * H_F * sizeof(float),
                 hipMemcpyDeviceToDevice, stream);
  {
    long long n = (long long)N_EDGES * (H_F / 4);
    gin_edge_agg<<<cdiv(n, BT), BT, 0, stream>>>(H1, src, dst, AGG1, N_EDGES, 4);
  }
  run_linear(AGG1, c1_w1, c1_b1, T2, N_NODES, H_F, H_F, 1, stream);
  run_linear(T2, c1_w2, c1_b2, H2, N_NODES, H_F, H_F, 1, stream);

  // ---- pooling ----
  hipMemsetAsync(SUM1, 0, (size_t)NGRAPH * H_F * sizeof(float), stream);
  hipMemsetAsync(SUM2, 0, (size_t)NGRAPH * H_F * sizeof(float), stream);
  hipMemsetAsync(CNT, 0, (size_t)NGRAPH * sizeof(float), stream);
  {
    long long n = (long long)N_NODES * H_F;
    pool_accum<<<cdiv(n, BT), BT, 0, stream>>>(H1, gid, SUM1, CNT, N_NODES, 6, 1);
    pool_accum<<<cdiv(n, BT), BT, 0, stream>>>(H2, gid, SUM2, CNT, N_NODES, 6, 0);
  }
  finalize_hg<<<cdiv(NGRAPH * (H_F + EV_F), BT), BT, 0, stream>>>(SUM1, SUM2, CNT, ev, TOT0);

  // ---- linear tower ----
  run_linear(TOT0, lw[0], lb[0], TA, NGRAPH, 80, 256, 1, stream);
  run_linear(TA,   lw[1], lb[1], TB, NGRAPH, 256, 128, 1, stream);
  run_linear(TB,   lw[2], lb[2], TA, NGRAPH, 128, 128, 1, stream);
  run_linear(TA,   lw[3], lb[3], TB, NGRAPH, 128, 64, 1, stream);
  run_linear(TB,   lw[4], lb[4], TA, NGRAPH, 64, 64, 1, stream);
  run_linear(TA,   lw[5], lb[5], TB, NGRAPH, 64, 32, 1, stream);
  final_dot<<<cdiv(NGRAPH, BT), BT, 0, stream>>>(TB, lw[6], lb[6], out, NGRAPH, 32);
}